// CVAE_31756988186635
// MI455X (gfx1250) — compile-verified
//
#include <hip/hip_runtime.h>
#include <hip/hip_bf16.h>

typedef __attribute__((ext_vector_type(2))) float v2f;
typedef __attribute__((ext_vector_type(8))) float v8f;

#define NITEMS 50000
#define EMB    16
#define SLATE  10
#define HID    512
#define LAT    64
#define RESP   1024
#define BATCH  256
#define ROWS   (BATCH * SLATE)              // 2560
#define IN_ENC (SLATE * EMB + EMB + RESP)   // 1200
#define IN_DEC (LAT + EMB + RESP)           // 1104

// ---------------------------------------------------------------------------
// 1) Ragged mean-pool: user[b,:] = (indicator_row . emb) / count
//    HBM-bound on the 51.2MB user_repr read; emb rows served from L2.
// ---------------------------------------------------------------------------
__global__ __launch_bounds__(256)
void pool_kernel(const int* __restrict__ user_repr,
                 const float* __restrict__ emb,
                 float* __restrict__ user) {
  const int b   = blockIdx.x;
  const int tid = threadIdx.x;
  float acc[EMB];
#pragma unroll
  for (int e = 0; e < EMB; ++e) acc[e] = 0.f;
  float cnt = 0.f;
  const int* ur = user_repr + (size_t)b * NITEMS;
  for (int n = tid; n < NITEMS; n += 256) {
    const float u = (float)ur[n];                 // 0 or 1, branch-free
    const float4* er = (const float4*)(emb + (size_t)n * EMB);
    float4 e0 = er[0], e1 = er[1], e2 = er[2], e3 = er[3];
    acc[0]  += u * e0.x; acc[1]  += u * e0.y; acc[2]  += u * e0.z; acc[3]  += u * e0.w;
    acc[4]  += u * e1.x; acc[5]  += u * e1.y; acc[6]  += u * e1.z; acc[7]  += u * e1.w;
    acc[8]  += u * e2.x; acc[9]  += u * e2.y; acc[10] += u * e2.z; acc[11] += u * e2.w;
    acc[12] += u * e3.x; acc[13] += u * e3.y; acc[14] += u * e3.z; acc[15] += u * e3.w;
    cnt += u;
  }
  __shared__ float red[EMB + 1][256];
#pragma unroll
  for (int e = 0; e < EMB; ++e) red[e][tid] = acc[e];
  red[EMB][tid] = cnt;
  __syncthreads();
  if (tid < EMB + 1) {
    float s = 0.f;
    for (int i = 0; i < 256; ++i) s += red[tid][i];
    red[tid][0] = s;
  }
  __syncthreads();
  if (tid < EMB) user[b * EMB + tid] = red[tid][0] / red[EMB][0];
}

// ---------------------------------------------------------------------------
// 2) Assemble encoder input x = [emb[slate], user, resp]   [B, 1200]
// ---------------------------------------------------------------------------
__global__ __launch_bounds__(256)
void prep_x_kernel(const int* __restrict__ slate,
                   const float* __restrict__ emb,
                   const float* __restrict__ user,
                   const float* __restrict__ resp,
                   float* __restrict__ x) {
  const int b = blockIdx.x;
  for (int i = threadIdx.x; i < IN_ENC; i += blockDim.x) {
    float v;
    if (i < SLATE * EMB) {
      const int k = i >> 4, e = i & 15;
      v = emb[(size_t)slate[b * SLATE + k] * EMB + e];
    } else if (i < SLATE * EMB + EMB) {
      v = user[b * EMB + (i - SLATE * EMB)];
    } else {
      v = resp[(size_t)b * RESP + (i - SLATE * EMB - EMB)];
    }
    x[(size_t)b * IN_ENC + i] = v;
  }
}

// ---------------------------------------------------------------------------
// 5) Assemble decoder input dz = [z, user, resp]           [B, 1104]
// ---------------------------------------------------------------------------
__global__ __launch_bounds__(256)
void prep_dz_kernel(const float* __restrict__ z,
                    const float* __restrict__ user,
                    const float* __restrict__ resp,
                    float* __restrict__ dz) {
  const int b = blockIdx.x;
  for (int i = threadIdx.x; i < IN_DEC; i += blockDim.x) {
    float v;
    if (i < LAT)            v = z[b * LAT + i];
    else if (i < LAT + EMB) v = user[b * EMB + (i - LAT)];
    else                    v = resp[(size_t)b * RESP + (i - LAT - EMB)];
    dz[(size_t)b * IN_DEC + i] = v;
  }
}

// ---------------------------------------------------------------------------
// Generic f32 WMMA GEMM: C[M,N] = act(A[M,K] @ W[N,K]^T + bias), K % 4 == 0.
// One wave per 16x16 C tile; V_WMMA_F32_16X16X4_F32 per K-chunk of 4, with
// two independent accumulators so consecutive WMMA chains interleave.
// A-operand: lanes 0-15 K={k,k+1}, lanes 16-31 K={k+2,k+3}; M = lane&15.
// B-operand: same K split; N = lane&15 (W row-major [N,K] -> contiguous).
// ---------------------------------------------------------------------------
__global__ __launch_bounds__(32)
void gemm_wmma_kernel(const float* __restrict__ A, int lda,
                      const float* __restrict__ W, int ldw,
                      const float* __restrict__ bias,
                      float* __restrict__ C, int ldc,
                      int Kdim, int relu) {
  const int lane = threadIdx.x;
  const int half = lane >> 4;
  const int l15  = lane & 15;
  const int koff = half * 2;
  const int tn = blockIdx.x, tm = blockIdx.y;
  const float* arow = A + (size_t)(tm * 16 + l15) * lda + koff;
  const float* wrow = W + (size_t)(tn * 16 + l15) * ldw + koff;
  v8f acc0 = {}, acc1 = {};
  int k = 0;
  for (; k + 8 <= Kdim; k += 8) {
    v2f a0, b0, a1, b1;
    a0.x = arow[k];     a0.y = arow[k + 1];
    b0.x = wrow[k];     b0.y = wrow[k + 1];
    a1.x = arow[k + 4]; a1.y = arow[k + 5];
    b1.x = wrow[k + 4]; b1.y = wrow[k + 5];
    acc0 = __builtin_amdgcn_wmma_f32_16x16x4_f32(
        false, a0, false, b0, (short)0, acc0, false, false);
    acc1 = __builtin_amdgcn_wmma_f32_16x16x4_f32(
        false, a1, false, b1, (short)0, acc1, false, false);
  }
  for (; k < Kdim; k += 4) {
    v2f a0, b0;
    a0.x = arow[k]; a0.y = arow[k + 1];
    b0.x = wrow[k]; b0.y = wrow[k + 1];
    acc0 = __builtin_amdgcn_wmma_f32_16x16x4_f32(
        false, a0, false, b0, (short)0, acc0, false, false);
  }
#pragma unroll
  for (int v = 0; v < 8; ++v) {
    const int row = tm * 16 + half * 8 + v;
    const int col = tn * 16 + l15;
    float r = acc0[v] + acc1[v] + bias[col];
    if (relu) r = fmaxf(r, 0.f);
    C[(size_t)row * ldc + col] = r;
  }
}

// ---------------------------------------------------------------------------
// 4) Reparameterized sample: z = z_mean + eps * exp(0.5*z_log_var)
// ---------------------------------------------------------------------------
__global__ __launch_bounds__(256)
void z_kernel(const float* __restrict__ zm, const float* __restrict__ zlv,
              const float* __restrict__ eps, float* __restrict__ z) {
  const int i = blockIdx.x * blockDim.x + threadIdx.x;
  if (i < BATCH * LAT) z[i] = zm[i] + eps[i] * __expf(0.5f * zlv[i]);
}

// ---------------------------------------------------------------------------
// 8) Scoring: logits[r, n] = rx[r,:16] . emb[n,:16], online softmax over n.
// One block per 16-row group; 16 waves stride over 3125 item tiles of 16.
// Each tile: 2x2 wmma_f32_16x16x4 (two independent chains) -> 16x16 logits,
// then a ONE-exp-per-logit online (max, sumexp, argmax) update:
//   exp(-|l-m|) is both the rescale factor (l>m) and the new term (l<=m).
// Half-wave shfl_xor reduce, cross-wave merge in LDS.
// Final prob at slate index = exp(l_slate - M)/S.
// ---------------------------------------------------------------------------
__global__ __launch_bounds__(512)
void score_kernel(const float* __restrict__ rx,    // [2560, 16]
                  const float* __restrict__ emb,   // [50000, 16]
                  const int* __restrict__ slate,   // [256, 10]
                  float* __restrict__ out) {
  const int lane = threadIdx.x & 31;
  const int wave = threadIdx.x >> 5;   // 0..15
  const int half = lane >> 4;
  const int l15  = lane & 15;
  const int koff = half * 2;
  const int rowg = blockIdx.x;         // 0..159
  const int row  = rowg * 16 + l15;

  // Preload A operand (this wave's 16 rx rows), constant across all tiles.
  v2f a[4];
#pragma unroll
  for (int c = 0; c < 4; ++c) {
    a[c].x = rx[(size_t)row * 16 + 4 * c + koff];
    a[c].y = rx[(size_t)row * 16 + 4 * c + koff + 1];
  }

  float m[8], s[8]; int idx[8];
#pragma unroll
  for (int v = 0; v < 8; ++v) { m[v] = -3.0e38f; s[v] = 0.f; idx[v] = 0; }

  const int NT = NITEMS / 16;          // 3125
#pragma unroll 2
  for (int t = wave; t < NT; t += 16) {
    const int n = t * 16 + l15;
    const float* er = emb + (size_t)n * 16 + koff;
    if (t + 16 < NT)
      __builtin_prefetch(emb + (size_t)(n + 256) * 16 + koff, 0, 3);
    v2f b0, b1, b2, b3;
    b0.x = er[0];  b0.y = er[1];
    b1.x = er[4];  b1.y = er[5];
    b2.x = er[8];  b2.y = er[9];
    b3.x = er[12]; b3.y = er[13];
    v8f acc0 = {}, acc1 = {};
    acc0 = __builtin_amdgcn_wmma_f32_16x16x4_f32(false, a[0], false, b0, (short)0, acc0, false, false);
    acc1 = __builtin_amdgcn_wmma_f32_16x16x4_f32(false, a[2], false, b2, (short)0, acc1, false, false);
    acc0 = __builtin_amdgcn_wmma_f32_16x16x4_f32(false, a[1], false, b1, (short)0, acc0, false, false);
    acc1 = __builtin_amdgcn_wmma_f32_16x16x4_f32(false, a[3], false, b3, (short)0, acc1, false, false);
#pragma unroll
    for (int v = 0; v < 8; ++v) {
      const float l = acc0[v] + acc1[v];     // row = half*8+v, item = n
      const float d = l - m[v];
      const float p = __expf(-fabsf(d));     // single exp per logit
      const bool gt = d > 0.f;
      s[v]   = gt ? fmaf(s[v], p, 1.f) : (s[v] + p);
      m[v]   = gt ? l : m[v];
      idx[v] = gt ? n : idx[v];
    }
  }

  // Reduce across the 16 lanes of each half (they share the same 8 rows).
#pragma unroll
  for (int off = 8; off >= 1; off >>= 1) {
#pragma unroll
    for (int v = 0; v < 8; ++v) {
      const float mo = __shfl_xor(m[v], off, 32);
      const float so = __shfl_xor(s[v], off, 32);
      const int   io = __shfl_xor(idx[v], off, 32);
      const float d  = mo - m[v];
      const float p  = __expf(-fabsf(d));
      const bool gt  = d > 0.f;
      s[v]   = gt ? fmaf(s[v], p, so) : fmaf(so, p, s[v]);
      m[v]   = gt ? mo : m[v];
      idx[v] = gt ? io : idx[v];
    }
  }

  __shared__ float lm[16][16], ls[16][16];
  __shared__ int   li[16][16];
  if (l15 == 0) {
#pragma unroll
    for (int v = 0; v < 8; ++v) {
      const int rl = half * 8 + v;
      lm[rl][wave] = m[v];
      ls[rl][wave] = s[v];
      li[rl][wave] = idx[v];
    }
  }
  __syncthreads();

  if (threadIdx.x < 16) {
    const int rl = threadIdx.x;
    float M = lm[rl][0], S = ls[rl][0]; int I = li[rl][0];
    for (int w = 1; w < 16; ++w) {
      const float mo = lm[rl][w], so = ls[rl][w]; const int io = li[rl][w];
      const float d  = mo - M;
      const float p  = __expf(-fabsf(d));
      const bool gt  = d > 0.f;
      S = gt ? fmaf(S, p, so) : fmaf(so, p, S);
      M = gt ? mo : M;
      I = gt ? io : I;
    }
    const int R = rowg * 16 + rl;
    const int b = R / SLATE, k = R % SLATE;
    const int si = slate[b * SLATE + k];
    const float* rr = rx + (size_t)R * 16;
    const float* ee = emb + (size_t)si * 16;
    float lg = 0.f;
#pragma unroll
    for (int e = 0; e < 16; ++e) lg += rr[e] * ee[e];
    out[2 * BATCH * LAT + R]        = (float)I;            // recon_slate
    out[2 * BATCH * LAT + ROWS + R] = __expf(lg - M) / S;  // recon_resp
  }
}

// ---------------------------------------------------------------------------
extern "C" void kernel_launch(void* const* d_in, const int* in_sizes, int n_in,
                              void* d_out, int out_size, void* d_ws, size_t ws_size,
                              hipStream_t stream) {
  (void)in_sizes; (void)n_in; (void)out_size; (void)ws_size;
  const int*   user_repr = (const int*)d_in[0];
  const int*   slate     = (const int*)d_in[1];
  const float* resp      = (const float*)d_in[2];
  const float* eps       = (const float*)d_in[3];
  const float* emb       = (const float*)d_in[4];
  const float* W_enc = (const float*)d_in[5];
  const float* b_enc = (const float*)d_in[6];
  const float* W_mu  = (const float*)d_in[7];
  const float* b_mu  = (const float*)d_in[8];
  const float* W_lv  = (const float*)d_in[9];
  const float* b_lv  = (const float*)d_in[10];
  const float* W_d1  = (const float*)d_in[11];
  const float* b_d1  = (const float*)d_in[12];
  const float* W_d2  = (const float*)d_in[13];
  const float* b_d2  = (const float*)d_in[14];

  float* out = (float*)d_out;
  float* ws  = (float*)d_ws;
  float* u_  = ws;                      // 256*16     =   4096
  float* x_  = u_  + BATCH * EMB;       // 256*1200   = 307200
  float* h_  = x_  + BATCH * IN_ENC;    // 256*512    = 131072
  float* z_  = h_  + BATCH * HID;       // 256*64     =  16384
  float* dz_ = z_  + BATCH * LAT;       // 256*1104   = 282624
  float* dh_ = dz_ + BATCH * IN_DEC;    // 256*512    = 131072
  float* rx_ = dh_ + BATCH * HID;       // 256*160    =  40960

  pool_kernel<<<BATCH, 256, 0, stream>>>(user_repr, emb, u_);
  prep_x_kernel<<<BATCH, 256, 0, stream>>>(slate, emb, u_, resp, x_);
  gemm_wmma_kernel<<<dim3(HID / 16, BATCH / 16), 32, 0, stream>>>(
      x_, IN_ENC, W_enc, IN_ENC, b_enc, h_, HID, IN_ENC, 1);
  gemm_wmma_kernel<<<dim3(LAT / 16, BATCH / 16), 32, 0, stream>>>(
      h_, HID, W_mu, HID, b_mu, out, LAT, HID, 0);                    // z_mean
  gemm_wmma_kernel<<<dim3(LAT / 16, BATCH / 16), 32, 0, stream>>>(
      h_, HID, W_lv, HID, b_lv, out + BATCH * LAT, LAT, HID, 0);      // z_log_var
  z_kernel<<<(BATCH * LAT + 255) / 256, 256, 0, stream>>>(
      out, out + BATCH * LAT, eps, z_);
  prep_dz_kernel<<<BATCH, 256, 0, stream>>>(z_, u_, resp, dz_);
  gemm_wmma_kernel<<<dim3(HID / 16, BATCH / 16), 32, 0, stream>>>(
      dz_, IN_DEC, W_d1, IN_DEC, b_d1, dh_, HID, IN_DEC, 1);
  gemm_wmma_kernel<<<dim3((SLATE * EMB) / 16, BATCH / 16), 32, 0, stream>>>(
      dh_, HID, W_d2, HID, b_d2, rx_, SLATE * EMB, HID, 1);
  score_kernel<<<ROWS / 16, 512, 0, stream>>>(rx_, emb, slate, out);
}